// ConstraintLoss_7215545058065
// MI455X (gfx1250) — compile-verified
//
#include <hip/hip_runtime.h>

typedef __attribute__((ext_vector_type(2))) float v2f;
typedef __attribute__((ext_vector_type(8))) float v8f;

#define G_FLOOR 0.02f
#define D_FLOOR 0.08f

__device__ __forceinline__ float wave_sum(float v) {
#pragma unroll
  for (int off = 16; off > 0; off >>= 1) v += __shfl_down(v, off, 32);
  return v;
}

__global__ void zero_ws_kernel(float* __restrict__ ws) {
  if (threadIdx.x < 8) ws[threadIdx.x] = 0.f;
}

// Block reduce of 4 running sums (8 waves of 256-thread block) using a single
// V_WMMA_F32_16X16X4_F32: B holds the 8x4 wave partials as a 4x16 matrix,
// A = ones, so D[m][n] = column-sum(n). Results atomically added to ws[0..3].
__device__ __forceinline__ void block_reduce4_wmma(float rx, float ry, float cx, float cy,
                                                   float* __restrict__ ws) {
  rx = wave_sum(rx);
  ry = wave_sum(ry);
  cx = wave_sum(cx);
  cy = wave_sum(cy);
  __shared__ float smem[8][4];
  const int wave = threadIdx.x >> 5;
  const int lane = threadIdx.x & 31;
  if (lane == 0) {
    smem[wave][0] = rx;
    smem[wave][1] = ry;
    smem[wave][2] = cx;
    smem[wave][3] = cy;
  }
  __syncthreads();
  if (wave == 0) {
#if __has_builtin(__builtin_amdgcn_wmma_f32_16x16x4_f32)
    // B (4x16 f32): VGPR0 -> rows {0,1} (lane halves), VGPR1 -> rows {2,3}.
    // Wave w maps to row k = w&3, column group grp = w>>2 (cols grp*4+q).
    const int hlf = lane >> 4;  // 0: lanes 0-15, 1: lanes 16-31
    const int nn  = lane & 15;  // column N
    float b0 = 0.f, b1 = 0.f;
    if (nn < 8) {
      const int q   = nn & 3;   // which quantity (rx,ry,cx,cy)
      const int grp = nn >> 2;  // wave group 0..3 vs 4..7
      b0 = smem[grp * 4 + hlf][q];      // row k = hlf
      b1 = smem[grp * 4 + 2 + hlf][q];  // row k = 2 + hlf
    }
    v2f A;  A[0] = 1.f; A[1] = 1.f;     // ones matrix
    v2f Bm; Bm[0] = b0; Bm[1] = b1;
    v8f C = {};
    v8f D = __builtin_amdgcn_wmma_f32_16x16x4_f32(
        /*neg_a=*/false, A, /*neg_b=*/false, Bm,
        /*c_mod=*/(short)0, C, /*reuse_a=*/false, /*reuse_b=*/false);
    float d0 = D[0];                     // D[0][n] = colsum(n) for lanes 0-15
    float other = __shfl(d0, (lane + 4) & 31, 32);
    if (lane < 4) atomicAdd(&ws[lane], d0 + other);
#else
    if (lane < 4) {
      float s = 0.f;
#pragma unroll
      for (int w = 0; w < 8; ++w) s += smem[w][lane];
      atomicAdd(&ws[lane], s);
    }
#endif
  }
}

__global__ __launch_bounds__(256) void smooth_sum_kernel(const float2* __restrict__ g,
                                                         const int* __restrict__ gsize,
                                                         int total_elems,
                                                         float* __restrict__ ws) {
  const unsigned n    = (unsigned)(*gsize);
  const unsigned nm2  = n - 2u;
  const unsigned Bc   = (unsigned)total_elems / (n * n * 2u);
  const unsigned R1   = Bc * n;    // rows for row-direction terms (b*n + i)
  const unsigned R2   = Bc * nm2;  // row-triple bases for col-direction terms
  const unsigned nwaves  = (gridDim.x * blockDim.x) >> 5;
  const unsigned wave_id = (blockIdx.x * blockDim.x + threadIdx.x) >> 5;
  const unsigned lane    = threadIdx.x & 31u;

  float rx = 0.f, ry = 0.f, cx = 0.f, cy = 0.f;

  // Row-direction terms: wave per row r = b*n + i, base offset = r*n.
  // No integer division anywhere; lanes stride contiguous j.
  for (unsigned r = wave_id; r < R1; r += nwaves) {
    const float2* base = g + (size_t)r * n;
    for (unsigned j = lane; j < nm2; j += 32u) {
      const float2 p0 = base[j];
      const float2 p1 = base[j + 1u];
      const float2 p2 = base[j + 2u];
      const float ax = p1.x - p0.x, bx = p2.x - p1.x;
      const float ay = p1.y - p0.y, by = p2.y - p1.y;
      rx += fmaxf(D_FLOOR, fabsf(bx * bx - ax * ax));
      ry += fmaxf(D_FLOOR, fabsf(by * by - ay * ay));
    }
  }

  // Col-direction terms: wave per triple-base row r2 = b*nm2 + i (i in [0, n-2)).
  // One division per row, amortized over n/32 lane-iterations.
  for (unsigned r = wave_id; r < R2; r += nwaves) {
    const unsigned b = r / nm2;
    const unsigned i = r - b * nm2;
    const float2* base = g + ((size_t)b * n + i) * n;
    for (unsigned j = lane; j < n; j += 32u) {
      const float2 p0 = base[j];
      const float2 p1 = base[j + n];
      const float2 p2 = base[j + 2u * n];
      const float ax = p1.x - p0.x, bx = p2.x - p1.x;
      const float ay = p1.y - p0.y, by = p2.y - p1.y;
      cx += fmaxf(D_FLOOR, fabsf(bx * bx - ax * ax));
      cy += fmaxf(D_FLOOR, fabsf(by * by - ay * ay));
    }
  }

  block_reduce4_wmma(rx, ry, cx, cy, ws);
}

// Geometry terms on batch 0 only (tiny): rg over j-triples, cg over i-triples.
// Wave-per-row, no divisions.
__global__ __launch_bounds__(256) void geom_kernel(const float2* __restrict__ g,
                                                   const int* __restrict__ gsize,
                                                   float* __restrict__ ws) {
  const unsigned n   = (unsigned)(*gsize);
  const unsigned nm2 = n - 2u;
  const unsigned nwaves  = (gridDim.x * blockDim.x) >> 5;
  const unsigned wave_id = (blockIdx.x * blockDim.x + threadIdx.x) >> 5;
  const unsigned lane    = threadIdx.x & 31u;

  float rg = 0.f, cg = 0.f;

  for (unsigned i = wave_id; i < n; i += nwaves) {
    const float2* base = g + (size_t)i * n;
    for (unsigned j = lane; j < nm2; j += 32u) {
      const float2 p0 = base[j];
      const float2 p1 = base[j + 1u];
      const float2 p2 = base[j + 2u];
      rg += fabsf((p1.y - p0.y) * (p1.x - p2.x) - (p1.y - p2.y) * (p1.x - p0.x));
    }
  }
  for (unsigned i = wave_id; i < nm2; i += nwaves) {
    const float2* base = g + (size_t)i * n;
    for (unsigned j = lane; j < n; j += 32u) {
      const float2 p0 = base[j];
      const float2 p1 = base[j + n];
      const float2 p2 = base[j + 2u * n];
      cg += fabsf((p1.y - p0.y) * (p1.x - p2.x) - (p1.y - p2.y) * (p1.x - p0.x));
    }
  }

  rg = wave_sum(rg);
  cg = wave_sum(cg);
  if (lane == 0) {
    atomicAdd(&ws[4], rg);
    atomicAdd(&ws[5], cg);
  }
}

__global__ void finalize_kernel(const float* __restrict__ ws,
                                const int* __restrict__ gsize,
                                int total_elems,
                                float* __restrict__ out) {
  if (threadIdx.x == 0 && blockIdx.x == 0) {
    const unsigned n   = (unsigned)(*gsize);
    const unsigned nm2 = n - 2u;
    const unsigned Bc  = (unsigned)total_elems / (n * n * 2u);
    const float inv = 1.0f / ((float)Bc * (float)n * (float)nm2);
    const float mean_part = (ws[0] + ws[1] + ws[2] + ws[3]) * inv;
    out[0] = mean_part + fmaxf(ws[4], G_FLOOR) + fmaxf(ws[5], G_FLOOR);
  }
}

extern "C" void kernel_launch(void* const* d_in, const int* in_sizes, int n_in,
                              void* d_out, int out_size, void* d_ws, size_t ws_size,
                              hipStream_t stream) {
  const float2* g     = (const float2*)d_in[0];
  const int*    gsize = (const int*)d_in[1];
  float*        ws    = (float*)d_ws;
  float*        out   = (float*)d_out;
  const int total = in_sizes[0];

  zero_ws_kernel<<<1, 32, 0, stream>>>(ws);
  smooth_sum_kernel<<<2048, 256, 0, stream>>>(g, gsize, total, ws);
  geom_kernel<<<64, 256, 0, stream>>>(g, gsize, ws);
  finalize_kernel<<<1, 32, 0, stream>>>(ws, gsize, total, out);
}